// myGCN_10514079941365
// MI455X (gfx1250) — compile-verified
//
#include <hip/hip_runtime.h>
#include <hip/hip_bf16.h>

// ---------------------------------------------------------------------------
// GCN (2x GCNConv + BatchNorm + ReLU) + linear head + sigmoid for gfx1250.
// N=100000 nodes, E=1600000 edges, D=128.
//
// Design notes (MI455X):
//  * fp32 WMMA (v_wmma_f32_16x16x4_f32) for both 128x128 GEMMs -> exact fp32.
//  * Weight matrix (64KB) staged into LDS via GLOBAL_LOAD_ASYNC_TO_LDS_B128
//    (CDNA5 async DMA, ASYNCcnt) while the A tile is staged with fused BN.
//  * 51.2MB feature matrix is L2-resident (192MB L2): edge gather + f32
//    atomic scatter run out of L2, not HBM.
//  * conv bias cancels inside BatchNorm -> dropped.
//  * BN-apply + ReLU fused into the next GEMM's LDS A-staging (scale/shift).
//  * t pre-scaled by dinv[src] so per-edge work = load*dinv[dst] + atomics.
// ---------------------------------------------------------------------------

typedef __attribute__((ext_vector_type(2))) float v2f;
typedef __attribute__((ext_vector_type(8))) float v8f;

#define DF 128
#define SA_STRIDE 132   // padded LDS row stride (floats) to avoid bank conflicts

// ---- degree / norm --------------------------------------------------------

__global__ void gcn_init_kernel(float* deg, float* scale, float* shift, int N) {
    int i = blockIdx.x * blockDim.x + threadIdx.x;
    if (i < N) deg[i] = 1.0f;            // self-loop
    if (i < DF) { scale[i] = 1.0f; shift[i] = 0.0f; }  // identity BN for layer 1
}

__global__ void gcn_degree_kernel(const int* __restrict__ dst, float* deg, int E) {
    int e = blockIdx.x * blockDim.x + threadIdx.x;
    if (e < E) atomicAdd(&deg[dst[e]], 1.0f);
}

__global__ void gcn_rsqrt_kernel(float* deg, int N) {
    int i = blockIdx.x * blockDim.x + threadIdx.x;
    if (i < N) deg[i] = rsqrtf(deg[i]);   // deg >= 1 always (self-loop)
}

// ---- WMMA GEMM: out[N,128] = f(A)[N,128] @ W[128,128] ---------------------
// f(a) = relu? max(0, a*scale[c]+shift[c]) : a*scale[c]+shift[c]
// One block = 16 output rows; 8 waves each own a 16-wide column tile.

__global__ __launch_bounds__(256) void gcn_gemm_kernel(
    const float* __restrict__ A, const float* __restrict__ W,
    const float* __restrict__ scale, const float* __restrict__ shift,
    float* __restrict__ out, int N, int applyRelu)
{
    __shared__ float sW[DF * DF];          // 64 KB: full weight matrix
    __shared__ float sA[16 * SA_STRIDE];   // ~8 KB: A tile (BN/ReLU applied)

    const int tid = threadIdx.x;
    const int rowBase = blockIdx.x * 16;

    // Stage W with CDNA5 async LDS DMA: 4096 x b128, 16 per thread.
    // Generic LDS pointer: addr[63:32] = shared aperture, addr[31:0] = offset.
    {
        const unsigned ldsBase = (unsigned)(size_t)(&sW[0]);
        const char* gBase = (const char*)W;
        for (int i = tid; i < DF * DF / 4; i += 256) {
            unsigned laddr = ldsBase + (unsigned)(i * 16);
            unsigned long long gaddr = (unsigned long long)(size_t)(gBase + (size_t)i * 16);
            asm volatile("global_load_async_to_lds_b128 %0, %1, off"
                         :: "v"(laddr), "v"(gaddr) : "memory");
        }
    }

    // Stage A tile with fused BN scale/shift (+ optional ReLU) while W lands.
    for (int i = tid; i < 16 * DF; i += 256) {
        int r = i >> 7, c = i & (DF - 1);
        int rr = rowBase + r;
        float v = (rr < N) ? A[(size_t)rr * DF + c] : 0.0f;
        v = fmaf(v, scale[c], shift[c]);
        if (applyRelu) v = fmaxf(v, 0.0f);
        sA[r * SA_STRIDE + c] = v;
    }

    asm volatile("s_wait_asynccnt 0" ::: "memory");
    __syncthreads();

    const int wave = tid >> 5;          // 0..7 -> column tile
    const int lane = tid & 31;
    const int colBase = wave * 16;
    const int m = lane & 15;            // row (A) / col (B) index within tile
    const int khalf = (lane >> 4) * 2;  // lanes 16-31 carry K+2/K+3

    v8f acc = {};
    #pragma unroll
    for (int k = 0; k < DF; k += 4) {
        const int kk = k + khalf;
        v2f a, b;
        a.x = sA[m * SA_STRIDE + kk];
        a.y = sA[m * SA_STRIDE + kk + 1];
        b.x = sW[kk * DF + colBase + m];
        b.y = sW[(kk + 1) * DF + colBase + m];
        acc = __builtin_amdgcn_wmma_f32_16x16x4_f32(
            /*neg_a=*/false, a, /*neg_b=*/false, b,
            /*c_mod=*/(short)0, acc, /*reuse_a=*/false, /*reuse_b=*/false);
    }

    // C/D layout: VGPR i -> rows i (lanes 0-15) and i+8 (lanes 16-31)
    const int mOff = (lane < 16) ? 0 : 8;
    const int n = lane & 15;
    #pragma unroll
    for (int i = 0; i < 8; ++i) {
        int rr = rowBase + mOff + i;
        if (rr < N) out[(size_t)rr * DF + colBase + n] = acc[i];
    }
}

// ---- prescale + self-loop init: t[i]*=dinv[i]; h[i] = t[i]*dinv[i] --------

__global__ void gcn_prescale_kernel(float* __restrict__ t,
                                    const float* __restrict__ dinv,
                                    float* __restrict__ h, int N)
{
    long i = (long)blockIdx.x * blockDim.x + threadIdx.x;
    if (i >= (long)N * DF) return;
    float di = dinv[i >> 7];
    float v = t[i] * di;
    t[i] = v;
    h[i] = v * di;   // self-loop contribution = h0 * dinv^2
}

// ---- edge scatter: wave(32 lanes) per edge, float4 per lane ---------------

__global__ __launch_bounds__(256) void gcn_scatter_kernel(
    const float* __restrict__ t, const int* __restrict__ src,
    const int* __restrict__ dst, const float* __restrict__ dinv,
    float* __restrict__ h, int E)
{
    int e = (int)(((long)blockIdx.x * blockDim.x + threadIdx.x) >> 5);
    if (e >= E) return;
    const int lane = threadIdx.x & 31;
    const int s = src[e];
    const int d = dst[e];
    const float w = dinv[d];               // t already scaled by dinv[s]
    const float4 v = ((const float4*)(t + (size_t)s * DF))[lane];
    float* hd = h + (size_t)d * DF + lane * 4;
    atomicAdd(hd + 0, v.x * w);
    atomicAdd(hd + 1, v.y * w);
    atomicAdd(hd + 2, v.z * w);
    atomicAdd(hd + 3, v.w * w);
}

// ---- batchnorm statistics --------------------------------------------------

__global__ void gcn_zero_stats_kernel(float* stats) {
    int i = threadIdx.x;
    if (i < 2 * DF) stats[i] = 0.0f;
}

__global__ __launch_bounds__(128) void gcn_stats_kernel(
    const float* __restrict__ h, float* __restrict__ stats, int N)
{
    const int col = threadIdx.x;   // 0..127
    float s = 0.0f, q = 0.0f;
    for (int r = blockIdx.x; r < N; r += gridDim.x) {
        float v = h[(size_t)r * DF + col];
        s += v;
        q += v * v;
    }
    atomicAdd(&stats[col], s);
    atomicAdd(&stats[DF + col], q);
}

__global__ void gcn_finalize_kernel(const float* __restrict__ stats,
                                    const float* __restrict__ gamma,
                                    const float* __restrict__ beta,
                                    float* scale, float* shift, float invN)
{
    int j = threadIdx.x;
    if (j < DF) {
        float mean = stats[j] * invN;
        float var = stats[DF + j] * invN - mean * mean;
        float sc = gamma[j] * rsqrtf(var + 1e-5f);
        scale[j] = sc;
        shift[j] = beta[j] - mean * sc;
    }
}

// ---- head: out = sigmoid(relu( relu(BN2(h)) @ Wl + bl )) ------------------

__global__ __launch_bounds__(256) void gcn_head_kernel(
    const float* __restrict__ h, const float* __restrict__ scale,
    const float* __restrict__ shift, const float* __restrict__ Wl,
    const float* __restrict__ bl, float* __restrict__ out, int N)
{
    int node = (int)(((long)blockIdx.x * blockDim.x + threadIdx.x) >> 5);
    if (node >= N) return;
    const int lane = threadIdx.x & 31;
    const int c = lane * 4;
    const float4 v = ((const float4*)(h + (size_t)node * DF))[lane];
    float p = 0.0f;
    p += fmaxf(fmaf(v.x, scale[c + 0], shift[c + 0]), 0.0f) * Wl[c + 0];
    p += fmaxf(fmaf(v.y, scale[c + 1], shift[c + 1]), 0.0f) * Wl[c + 1];
    p += fmaxf(fmaf(v.z, scale[c + 2], shift[c + 2]), 0.0f) * Wl[c + 2];
    p += fmaxf(fmaf(v.w, scale[c + 3], shift[c + 3]), 0.0f) * Wl[c + 3];
    #pragma unroll
    for (int off = 16; off > 0; off >>= 1) p += __shfl_down(p, off, 32);
    if (lane == 0) {
        float y = fmaxf(p + bl[0], 0.0f);
        out[node] = 1.0f / (1.0f + __expf(-y));
    }
}

// ---------------------------------------------------------------------------

static inline size_t alignUp(size_t x) { return (x + 255) & ~(size_t)255; }

extern "C" void kernel_launch(void* const* d_in, const int* in_sizes, int n_in,
                              void* d_out, int out_size, void* d_ws, size_t ws_size,
                              hipStream_t stream) {
    const float* x   = (const float*)d_in[0];
    const int*   ei  = (const int*)d_in[1];
    const float* W1  = (const float*)d_in[2];
    // d_in[3] = b1: cancels inside BatchNorm
    const float* g1  = (const float*)d_in[4];
    const float* be1 = (const float*)d_in[5];
    const float* W2  = (const float*)d_in[6];
    // d_in[7] = b2: cancels inside BatchNorm
    const float* g2  = (const float*)d_in[8];
    const float* be2 = (const float*)d_in[9];
    const float* Wl  = (const float*)d_in[10];
    const float* bl  = (const float*)d_in[11];
    float* outp = (float*)d_out;

    const int N = in_sizes[0] / DF;
    const int E = in_sizes[1] / 2;
    const int* src = ei;
    const int* dst = ei + E;

    // workspace layout
    char* ws = (char*)d_ws;
    size_t off = 0;
    float* dinv  = (float*)(ws + off); off += alignUp((size_t)N * sizeof(float));
    float* t     = (float*)(ws + off); off += alignUp((size_t)N * DF * sizeof(float));
    float* h     = (float*)(ws + off); off += alignUp((size_t)N * DF * sizeof(float));
    float* stats = (float*)(ws + off); off += alignUp(2 * DF * sizeof(float));
    float* scale = (float*)(ws + off); off += alignUp(DF * sizeof(float));
    float* shift = (float*)(ws + off); off += alignUp(DF * sizeof(float));
    (void)ws_size; (void)n_in; (void)out_size;

    const int bN    = (N + 255) / 256;
    const int bE    = (E + 255) / 256;
    const int bGemm = (N + 15) / 16;
    const int bElem = (int)(((long)N * DF + 255) / 256);
    const int bScat = (int)(((long)E * 32 + 255) / 256);
    const int bHead = (int)(((long)N * 32 + 255) / 256);
    const float invN = 1.0f / (float)N;

    // normalization (shared by both layers)
    gcn_init_kernel<<<bN, 256, 0, stream>>>(dinv, scale, shift, N);
    gcn_degree_kernel<<<bE, 256, 0, stream>>>(dst, dinv, E);
    gcn_rsqrt_kernel<<<bN, 256, 0, stream>>>(dinv, N);

    // ---- layer 1: t = x @ W1 ; h = scatter ; BN1 params -> scale/shift
    gcn_gemm_kernel<<<bGemm, 256, 0, stream>>>(x, W1, scale, shift, t, N, 0);
    gcn_prescale_kernel<<<bElem, 256, 0, stream>>>(t, dinv, h, N);
    gcn_scatter_kernel<<<bScat, 256, 0, stream>>>(t, src, dst, dinv, h, E);
    gcn_zero_stats_kernel<<<1, 256, 0, stream>>>(stats);
    gcn_stats_kernel<<<1024, 128, 0, stream>>>(h, stats, N);
    gcn_finalize_kernel<<<1, 128, 0, stream>>>(stats, g1, be1, scale, shift, invN);

    // ---- layer 2: t = relu(BN1(h)) @ W2 ; h = scatter ; BN2 -> scale/shift
    gcn_gemm_kernel<<<bGemm, 256, 0, stream>>>(h, W2, scale, shift, t, N, 1);
    gcn_prescale_kernel<<<bElem, 256, 0, stream>>>(t, dinv, h, N);
    gcn_scatter_kernel<<<bScat, 256, 0, stream>>>(t, src, dst, dinv, h, E);
    gcn_zero_stats_kernel<<<1, 256, 0, stream>>>(stats);
    gcn_stats_kernel<<<1024, 128, 0, stream>>>(h, stats, N);
    gcn_finalize_kernel<<<1, 128, 0, stream>>>(stats, g2, be2, scale, shift, invN);

    // ---- head
    gcn_head_kernel<<<bHead, 256, 0, stream>>>(h, scale, shift, Wl, bl, outp, N);
}